// R2NPropositionalLayer_11630771437757
// MI455X (gfx1250) — compile-verified
//
#include <hip/hip_runtime.h>

typedef unsigned int u32;
typedef unsigned long long u64;
typedef u32 v4u __attribute__((ext_vector_type(4)));
typedef int  v8i __attribute__((ext_vector_type(8)));
typedef int  v4i __attribute__((ext_vector_type(4)));

#define CC 64          // columns of x
#define RR 16          // rules
#define KK 4           // clauses per rule
#define MM 3           // literals per clause
#define ROWS 256       // rows per block (one per thread)
#define LDS_STRIDE 65  // 64 data dwords + 1 pad dword (TDM pad: interval 64 dw, amount 1 dw)

__global__ __launch_bounds__(ROWS) void r2n_prop_kernel(
    const float* __restrict__ x,
    const int* __restrict__ rule_idx,
    const unsigned char* __restrict__ rule_sign,
    float* __restrict__ out,
    int nrows)
{
    __shared__ float s_tile[ROWS * LDS_STRIDE + 4];

    const int tid       = threadIdx.x;
    const int row_base  = blockIdx.x * ROWS;
    const int rows_this = min(ROWS, nrows - row_base);

    // Plain LDS store of a thread-dependent value into this row's pad dword:
    // keeps s_tile a "written" object so the dynamic-index gathers below can't
    // be folded to undef. Pad dwords are skipped by the TDM and never gathered
    // (idx < 64), so this is semantically inert. (ds_store_b32, no wait.)
    s_tile[tid * LDS_STRIDE + CC] = (float)tid;

    if (tid < 32) {
        // ---- Tensor Data Mover: DMA 2-D tile (64 x rows_this fp32) global -> LDS ----
        // LDS padding inserts 1 dword after every 64 dwords => row stride 65 dwords,
        // which makes the per-thread gathers bank-conflict-free.
        u64 gaddr = (u64)(const void*)x + (u64)row_base * (CC * 4);

        v4u g0;
        g0.x = 1u;                                   // count=1 (valid user descriptor)
        g0.y = 0u;                                   // lds_addr = 0 (s_tile is sole LDS block)
        g0.z = (u32)(gaddr & 0xFFFFFFFFull);         // global_addr[31:0]
        g0.w = (u32)((gaddr >> 32) & 0x1FFFFFFull)   // global_addr[56:32]
             | (2u << 30);                           // type = 2 ("image")

        v8i g1;
        g1[0] = (2 << 16)                            // data_size = 4 bytes
              | (1 << 20)                            // pad_enable
              | (5 << 22)                            // pad_interval: every 64 dwords
              | (0 << 25);                           // pad_amount: 1 dword
        g1[1] = (CC & 0xFFFF) << 16;                 // tensor_dim0 = 64   (bits 79:48, lo half)
        g1[2] = (rows_this & 0xFFFF) << 16;          // dim0 hi=0 | tensor_dim1 lo = rows_this
        g1[3] = (CC << 16);                          // dim1 hi=0 | tile_dim0 = 64
        g1[4] = (rows_this & 0xFFFF);                // tile_dim1 = rows_this, tile_dim2 = 0
        g1[5] = CC;                                  // tensor_dim0_stride = 64 (lo 32)
        g1[6] = 0;                                   // stride0 hi | stride1 lo
        g1[7] = 0;                                   // stride1 hi

        v4i g2 = {0, 0, 0, 0};                       // no dims 2/3 (2-D tensor)
        v4i g3 = {0, 0, 0, 0};
        v8i g4 = {0, 0, 0, 0, 0, 0, 0, 0};           // extra group (6-arg builtin), zero

        __builtin_amdgcn_tensor_load_to_lds(g0, g1, g2, g3, g4, 0);
        __builtin_amdgcn_s_wait_tensorcnt(0);
    }
    __syncthreads();
    // Escape s_tile's address into the asm and clobber memory: the TDM wrote
    // LDS behind the compiler's back, so force it to treat the array contents
    // as unknown from here on.
    asm volatile("" :: "v"((u32)(uintptr_t)&s_tile[0]) : "memory");

    const int row = row_base + tid;
    if (row < nrows) {
        const float* __restrict__ lx = &s_tile[tid * LDS_STRIDE];
        float res[RR];
#pragma unroll
        for (int r = 0; r < RR; ++r) {
            float q = 1.0f;
#pragma unroll
            for (int k = 0; k < KK; ++k) {
                float c = 1.0f;
#pragma unroll
                for (int m = 0; m < MM; ++m) {
                    const int t  = (r * KK + k) * MM + m;
                    const float xi  = lx[rule_idx[t]];       // ds_load gather
                    const float lit = rule_sign[t] ? (1.0f - xi) : xi;
                    c *= lit;
                }
                q *= (1.0f - c);
            }
            res[r] = 1.0f - q;
        }
        // 16 consecutive floats per row -> 4x b128 stores, coalesced across lanes.
        float4* o = (float4*)(out + (size_t)row * RR);
        o[0] = make_float4(res[0],  res[1],  res[2],  res[3]);
        o[1] = make_float4(res[4],  res[5],  res[6],  res[7]);
        o[2] = make_float4(res[8],  res[9],  res[10], res[11]);
        o[3] = make_float4(res[12], res[13], res[14], res[15]);
    }
}

extern "C" void kernel_launch(void* const* d_in, const int* in_sizes, int n_in,
                              void* d_out, int out_size, void* d_ws, size_t ws_size,
                              hipStream_t stream) {
    const float*         x         = (const float*)d_in[0];
    const int*           rule_idx  = (const int*)d_in[1];
    const unsigned char* rule_sign = (const unsigned char*)d_in[2];
    float*               out       = (float*)d_out;

    const int nrows = in_sizes[0] / CC;   // 500000
    dim3 grid((nrows + ROWS - 1) / ROWS); // 1954 blocks
    hipLaunchKernelGGL(r2n_prop_kernel, grid, dim3(ROWS), 0, stream,
                       x, rule_idx, rule_sign, out, nrows);
}